// DiagramNet_47940424958188
// MI455X (gfx1250) — compile-verified
//
#include <hip/hip_runtime.h>

// Problem dims (fixed by the reference).
#define B_   64
#define LQ_  64
#define O_   8
#define LO_  32
#define S_   64
#define W_   32
#define D_   512
#define NEG_BIG (-9.0e15f)

typedef __attribute__((ext_vector_type(2))) float v2f;
typedef __attribute__((ext_vector_type(8))) float v8f;

// ---------------------------------------------------------------------------
// Kernel 1: csum[b,s,d] = sum_w csf[b,s,w,d]   and  csf_len[b,s]
// Dominant-bandwidth pass: streams all 268 MB of csf exactly once.
// One block per (b,s); 128 threads x float4 cover D=512; loop over W=32.
// ---------------------------------------------------------------------------
__global__ __launch_bounds__(128)
void k_csum(const float* __restrict__ csf,
            const unsigned char* __restrict__ csf_mask,
            float* __restrict__ csum,
            float* __restrict__ csf_len) {
  const int bs = blockIdx.x;          // b*S + s
  const int t  = threadIdx.x;         // 0..127 -> float4 slice of D
  const float* src = csf + (size_t)bs * (W_ * D_) + t * 4;
  float a0 = 0.f, a1 = 0.f, a2 = 0.f, a3 = 0.f;
  #pragma unroll 4
  for (int w = 0; w < W_; ++w) {
    const float4 v = *(const float4*)(src + (size_t)w * D_);
    a0 += v.x; a1 += v.y; a2 += v.z; a3 += v.w;
  }
  float4 out; out.x = a0; out.y = a1; out.z = a2; out.w = a3;
  *(float4*)(csum + (size_t)bs * D_ + t * 4) = out;

  if (t == 0) {
    const unsigned char* m = csf_mask + bs * W_;
    int cnt = 0;
    for (int w = 0; w < W_; ++w) cnt += (m[w] ? 0 : 1);
    csf_len[bs] = (cnt == 0) ? NEG_BIG : (float)cnt;
  }
}

// ---------------------------------------------------------------------------
// Kernel 2: qsum[b,d] = sum_q que[b,q,d];  osum[b,o,d] = sum_l opt[b,o,l,d]
// block = b*9 + r  (r==0 -> que row, r=1..8 -> option r-1)
// ---------------------------------------------------------------------------
__global__ __launch_bounds__(128)
void k_rowsums(const float* __restrict__ que,
               const float* __restrict__ opt,
               float* __restrict__ qsum,
               float* __restrict__ osum) {
  const int br = blockIdx.x;
  const int b = br / 9, r = br % 9, t = threadIdx.x;
  float a0 = 0.f, a1 = 0.f, a2 = 0.f, a3 = 0.f;
  if (r == 0) {
    const float* src = que + (size_t)b * LQ_ * D_ + t * 4;
    for (int q = 0; q < LQ_; ++q) {
      const float4 v = *(const float4*)(src + (size_t)q * D_);
      a0 += v.x; a1 += v.y; a2 += v.z; a3 += v.w;
    }
    float4 out; out.x = a0; out.y = a1; out.z = a2; out.w = a3;
    *(float4*)(qsum + (size_t)b * D_ + t * 4) = out;
  } else {
    const int o = r - 1;
    const float* src = opt + (size_t)(b * O_ + o) * LO_ * D_ + t * 4;
    for (int l = 0; l < LO_; ++l) {
      const float4 v = *(const float4*)(src + (size_t)l * D_);
      a0 += v.x; a1 += v.y; a2 += v.z; a3 += v.w;
    }
    float4 out; out.x = a0; out.y = a1; out.z = a2; out.w = a3;
    *(float4*)(osum + (size_t)(b * O_ + o) * D_ + t * 4) = out;
  }
}

// ---------------------------------------------------------------------------
// Kernel 3: scores + argmax via V_WMMA_F32_16X16X4_F32 (full f32 precision).
// One wave (32 lanes, wave32) per batch item.
//   A (16 x 512): row 0 = qsum[b], rows 1..8 = osum[b,o], rows 9..15 = 0
//   B (512 x 16): columns = csum[b, s0+n, :]  (4 N-tiles cover S=64)
//   D (16 x 16) accumulated over D_=512 in K=4 steps (128 WMMA per tile).
// f32 A-fragment layout (ISA 7.12.2): lane m=lane&15, lanes>=16 hold K+2/K+3.
// Scores go to LDS, then lanes 0..8 scan argmax over s of score/csf_len.
// ---------------------------------------------------------------------------
__global__ __launch_bounds__(32)
void k_scores(const float* __restrict__ qsum,
              const float* __restrict__ osum,
              const float* __restrict__ csum,
              const float* __restrict__ csf_len,
              int* __restrict__ ix) {
  __shared__ float sc[16][S_ + 1];    // +1 pad to dodge bank conflicts
  const int b    = blockIdx.x;
  const int lane = threadIdx.x;       // 0..31 (wave32)
  const int m    = lane & 15;
  const int koff = (lane >> 4) * 2;   // lanes 16..31 hold K+2, K+3

  // A-row pointer; rows > 8 read row 0 but are scaled to zero -> no lane
  // divergence, EXEC stays all-ones for every WMMA issue.
  const float* aptr = (m == 0) ? (qsum + (size_t)b * D_)
                    : (m <= 8) ? (osum + (size_t)(b * O_ + (m - 1)) * D_)
                               : (qsum + (size_t)b * D_);
  const float ascale = (m <= 8) ? 1.0f : 0.0f;

  for (int tile = 0; tile < 4; ++tile) {
    const int col = tile * 16 + m;                       // sentence index s
    const float* bptr = csum + (size_t)(b * S_ + col) * D_;
    v8f acc = {};
    for (int k = 0; k < D_; k += 4) {
      v2f av, bv;
      av.x = aptr[k + koff] * ascale;
      av.y = aptr[k + koff + 1] * ascale;
      bv.x = bptr[k + koff];
      bv.y = bptr[k + koff + 1];
      acc = __builtin_amdgcn_wmma_f32_16x16x4_f32(
          /*neg_a=*/false, av, /*neg_b=*/false, bv,
          /*c_mod=*/(short)0, acc, /*reuse_a=*/false, /*reuse_b=*/false);
    }
    // C/D layout: lanes 0-15 hold rows 0..7 in acc[0..7], lanes 16-31 rows 8..15
    const int rbase = (lane < 16) ? 0 : 8;
    #pragma unroll
    for (int v = 0; v < 8; ++v) sc[rbase + v][col] = acc[v];
  }
  __syncthreads();

  if (lane < 9) {                     // row 0: question; rows 1..8: options
    float best = -3.4e38f;
    int   bi   = 0;
    for (int s = 0; s < S_; ++s) {
      const float v = sc[lane][s] / csf_len[b * S_ + s];
      if (v > best) { best = v; bi = s; }   // strict '>' == first-max (argmax)
    }
    ix[b * 16 + lane] = bi;
  }
}

// ---------------------------------------------------------------------------
// Kernel 4: gather + outputs + masks. One block per (b,o,w) row pair.
//   que_csf[b,o,w,:] = opt_sum ? 0 : csf[b, ix_q2c[b], w, :]
//   opt_csf[b,o,w,:] = opt_sum ? 0 : csf[0, ix_o2c[b,o], w, :]   (batch 0!)
//   masks = (sum_d |row|) == 0, emitted as 1.0/0.0 floats.
// ---------------------------------------------------------------------------
__global__ __launch_bounds__(128)
void k_gather(const float* __restrict__ csf,
              const unsigned char* __restrict__ opt_sum,
              const int* __restrict__ ix,
              float* __restrict__ out_que,
              float* __restrict__ out_qmask,
              float* __restrict__ out_opt,
              float* __restrict__ out_omask) {
  __shared__ float redq[128];
  __shared__ float redo[128];
  const int row = blockIdx.x;         // (b*O + o)*W + w
  const int w   = row % W_;
  const int bo  = row / W_;
  const int o   = bo % O_;
  const int b   = bo / O_;
  const int t   = threadIdx.x;

  const int   ixq = ix[b * 16];
  const int   ixo = ix[b * 16 + 1 + o];
  const float zf  = opt_sum[b * O_ + o] ? 0.0f : 1.0f;

  const float* qsrc = csf + ((size_t)(b * S_ + ixq) * W_ + w) * D_ + t * 4;
  const float* osrc = csf + ((size_t)ixo * W_ + w) * D_ + t * 4;  // csf[0]

  float4 qv = *(const float4*)qsrc;
  float4 ov = *(const float4*)osrc;
  qv.x *= zf; qv.y *= zf; qv.z *= zf; qv.w *= zf;
  ov.x *= zf; ov.y *= zf; ov.z *= zf; ov.w *= zf;

  *(float4*)(out_que + (size_t)row * D_ + t * 4) = qv;
  *(float4*)(out_opt + (size_t)row * D_ + t * 4) = ov;

  redq[t] = fabsf(qv.x) + fabsf(qv.y) + fabsf(qv.z) + fabsf(qv.w);
  redo[t] = fabsf(ov.x) + fabsf(ov.y) + fabsf(ov.z) + fabsf(ov.w);
  __syncthreads();
  for (int off = 64; off > 0; off >>= 1) {
    if (t < off) { redq[t] += redq[t + off]; redo[t] += redo[t + off]; }
    __syncthreads();
  }
  if (t == 0) {
    out_qmask[row] = (redq[0] == 0.0f) ? 1.0f : 0.0f;
    out_omask[row] = (redo[0] == 0.0f) ? 1.0f : 0.0f;
  }
}

// ---------------------------------------------------------------------------
// Host-side launch.
// ---------------------------------------------------------------------------
extern "C" void kernel_launch(void* const* d_in, const int* in_sizes, int n_in,
                              void* d_out, int out_size, void* d_ws, size_t ws_size,
                              hipStream_t stream) {
  (void)in_sizes; (void)n_in; (void)out_size; (void)ws_size;

  const float*         que      = (const float*)d_in[0];          // [B,LQ,D]
  const float*         opt      = (const float*)d_in[1];          // [B,O,LO,D]
  const float*         csf      = (const float*)d_in[2];          // [B,S,W,D]
  const unsigned char* csf_mask = (const unsigned char*)d_in[3];  // [B,S,W] bool
  const unsigned char* opt_sum  = (const unsigned char*)d_in[4];  // [B,O]   bool

  // Workspace layout (floats): csum | qsum | osum | csf_len | ix(int)
  float* ws    = (float*)d_ws;
  float* csum  = ws;                                   // B*S*D  = 2,097,152
  float* qsum  = csum + (size_t)B_ * S_ * D_;          // B*D    =    32,768
  float* osum  = qsum + (size_t)B_ * D_;               // B*O*D  =   262,144
  float* clen  = osum + (size_t)B_ * O_ * D_;          // B*S    =     4,096
  int*   ix    = (int*)(clen + B_ * S_);               // B*16 ints

  // Output layout: que_csf | que_csf_mask | opt_csf | opt_csf_mask (all f32)
  float* out_que   = (float*)d_out;
  float* out_qmask = out_que   + (size_t)B_ * O_ * W_ * D_;
  float* out_opt   = out_qmask + (size_t)B_ * O_ * W_;
  float* out_omask = out_opt   + (size_t)B_ * O_ * W_ * D_;

  k_csum   <<<B_ * S_,       128, 0, stream>>>(csf, csf_mask, csum, clen);
  k_rowsums<<<B_ * 9,        128, 0, stream>>>(que, opt, qsum, osum);
  k_scores <<<B_,             32, 0, stream>>>(qsum, osum, csum, clen, ix);
  k_gather <<<B_ * O_ * W_,  128, 0, stream>>>(csf, opt_sum, ix,
                                               out_que, out_qmask,
                                               out_opt, out_omask);
}